// MultiHeadDotProductAttentionAqt_9981503995910
// MI455X (gfx1250) — compile-verified
//
#include <hip/hip_runtime.h>
#include <hip/hip_bf16.h>

typedef __attribute__((ext_vector_type(2))) float v2f;
typedef __attribute__((ext_vector_type(8))) float v8f;
typedef __attribute__((ext_vector_type(4))) unsigned int u32x4;
typedef __attribute__((ext_vector_type(8))) int i32x8;
typedef __attribute__((ext_vector_type(4))) int i32x4;

#define WMMA_F32(a, b, c) \
  __builtin_amdgcn_wmma_f32_16x16x4_f32(false, (a), false, (b), (short)0, (c), false, false)

#if __has_builtin(__builtin_amdgcn_tensor_load_to_lds) && \
    __has_builtin(__builtin_amdgcn_s_wait_tensorcnt)
#define USE_TDM 1
#else
#define USE_TDM 0
#endif

namespace {
constexpr int Bn = 4;
constexpr int Sn = 1024;
constexpr int Hn = 16;
constexpr int Dn = 64;
constexpr int NT = Sn / 16;                     // 64 key tiles
constexpr int ROWS = Hn * Dn;                   // floats between seq positions
constexpr float EXP_C = 0.000335462627902512f;  // exp(-8)
}  // namespace

#if USE_TDM
// Issue one TDM load of a 16-row x 64-float tile (row stride ROWS floats in
// global) into contiguous LDS at byte offset ldsOff.  D# per ISA 8.3/8.4:
// 2D tensor == tile, data_size=4B, type=2, groups 2/3 unused (zero).
__device__ __forceinline__ void tdm_load_tile(unsigned ldsOff,
                                              const float* gaddr) {
  unsigned long long ga = (unsigned long long)(uintptr_t)gaddr;
  u32x4 g0;
  g0[0] = 1u;  // count=1, user descriptor, no gather
  g0[1] = ldsOff;
  g0[2] = (unsigned)(ga & 0xFFFFFFFFu);
  g0[3] = (unsigned)((ga >> 32) & 0x01FFFFFFu) | (2u << 30);  // type=2
  i32x8 g1;
  g1[0] = 0x00020000;        // data_size=2 (4 bytes); no multicast/pad
  g1[1] = (int)(64u << 16);  // tensor_dim0[15:0] = 64
  g1[2] = (int)(16u << 16);  // tensor_dim1[15:0] = 16
  g1[3] = (int)(64u << 16);  // tile_dim0 = 64
  g1[4] = 16;                // tile_dim1 = 16, tile_dim2 = 0
  g1[5] = ROWS;              // tensor_dim0_stride = 1024 floats
  g1[6] = 0;
  g1[7] = 0;
  i32x4 z4 = {0, 0, 0, 0};
  i32x8 z8 = {0, 0, 0, 0, 0, 0, 0, 0};
  __builtin_amdgcn_tensor_load_to_lds(g0, g1, z4, z4, z8, 0);
}
#endif

// One 16x16 score tile  S^T = K_tile x Q^T  (D-layout result).
// base points at key-row 0 of the tile; rs = floats between key rows.
// Lane n holds the QUERY column; VGPR j holds key row (j + 8*hi).
__device__ __forceinline__ v8f score_tile(const float* base, int rs, int n,
                                          int hi, const v2f* bQ) {
  const float* krow = base + n * rs;
  v8f s = {0.f, 0.f, 0.f, 0.f, 0.f, 0.f, 0.f, 0.f};
#pragma unroll
  for (int c = 0; c < 16; ++c) {
    const int f = 4 * c + 2 * hi;
    v2f aK = *reinterpret_cast<const v2f*>(krow + f);  // 8B aligned
    s = WMMA_F32(aK, bQ[c], s);
  }
  return s;
}

__global__ __launch_bounds__(128) void attn_fp32_wmma(
    const float* __restrict__ q, const float* __restrict__ k,
    const float* __restrict__ v, float* __restrict__ out) {
  const int lane = threadIdx.x & 31;
  const int wave = threadIdx.x >> 5;
  const int n = lane & 15;   // query column (GEMM1) / d column (GEMM2)
  const int hi = lane >> 4;  // wave half
  const int h = blockIdx.y;
  const int b = blockIdx.z;
  const int qbase = (blockIdx.x * 4 + wave) * 16;

  const size_t headOff = (size_t)b * Sn * ROWS + (size_t)h * Dn;
  const float* qp = q + headOff;
  const float* kp = k + headOff;
  const float* vp = v + headOff;
  float* op = out + headOff;

#if USE_TDM
  // Dynamic LDS, triple-buffered: K tiles at buf*1024 floats (buf=0..2),
  // V tiles at 3072 + buf*1024 floats.  24 KB total.
  extern __shared__ float smem[];
#endif

  // ---- load Q tile as B-fragments of Q^T (scaled by 1/sqrt(D)) ----
  v2f bQ[16];
  {
    const float* qrow = qp + (size_t)(qbase + n) * ROWS;
#pragma unroll
    for (int c = 0; c < 16; ++c) {
      const int f = 4 * c + 2 * hi;
      v2f qv = *reinterpret_cast<const v2f*>(qrow + f);
      bQ[c].x = qv.x * 0.125f;
      bQ[c].y = qv.y * 0.125f;
    }
  }

  // ================= pass 1: exact row max over all keys =================
  // Pipeline: lookahead 2, 3 buffers, ONE barrier per tile.
  float halfmax = -3.0e38f;
#if USE_TDM
  if (wave == 0) {  // prologue: tiles 0,1 -> K bufs 0,1
    tdm_load_tile(0u, kp);
    tdm_load_tile(1024u * 4u, kp + (size_t)16 * ROWS);
  }
  int cur = 0;  // t % 3
  int pre = 2;  // (t+2) % 3
#endif
  for (int t = 0; t < NT; ++t) {
#if USE_TDM
    if (wave == 0) {
      if (t < NT - 1) {
        __builtin_amdgcn_s_wait_tensorcnt(1);  // tile t landed
      } else {
        __builtin_amdgcn_s_wait_tensorcnt(0);
      }
    }
    // One barrier: publishes tile t to all waves AND proves every wave is
    // done consuming tile t-1 (whose buffer tile t+2 will overwrite).
    __syncthreads();
    if (wave == 0 && t + 2 < NT) {
      tdm_load_tile((unsigned)(pre * 1024 * 4),
                    kp + (size_t)(t + 2) * 16 * ROWS);
    }
    const float* kt = smem + cur * 1024;
    const int krs = 64;
    cur = (cur == 2) ? 0 : cur + 1;
    pre = (pre == 2) ? 0 : pre + 1;
#else
    const float* kt = kp + (size_t)t * 16 * ROWS;
    const int krs = ROWS;
#endif
    v8f s = score_tile(kt, krs, n, hi, bQ);
#pragma unroll
    for (int j = 0; j < 8; ++j) halfmax = fmaxf(halfmax, s[j]);
  }
  float qmax = fmaxf(halfmax, __shfl_xor(halfmax, 16, 32));
  if (!__builtin_isfinite(qmax)) qmax = 0.0f;  // jnp.where(isfinite, amax, 0)

  // ============ pass 2: clipped exp, denominator, and P x V ==============
  float halfsum = 0.0f;
  v8f O0 = {0.f, 0.f, 0.f, 0.f, 0.f, 0.f, 0.f, 0.f};
  v8f O1 = O0, O2 = O0, O3 = O0;

#if USE_TDM
  __syncthreads();  // pass-1 consumption fully done before buffers reused
  if (wave == 0) {  // prologue: tiles 0,1 (K+V) -> bufs 0,1
    tdm_load_tile(0u, kp);
    tdm_load_tile((unsigned)(3072 * 4), vp);
    tdm_load_tile((unsigned)(1024 * 4), kp + (size_t)16 * ROWS);
    tdm_load_tile((unsigned)((3072 + 1024) * 4), vp + (size_t)16 * ROWS);
  }
  int cur2 = 0;
  int pre2 = 2;
#endif
  for (int t = 0; t < NT; ++t) {
#if USE_TDM
    if (wave == 0) {
      if (t < NT - 1) {
        __builtin_amdgcn_s_wait_tensorcnt(2);  // tile t's K+V landed
      } else {
        __builtin_amdgcn_s_wait_tensorcnt(0);
      }
    }
    __syncthreads();
    if (wave == 0 && t + 2 < NT) {
      const size_t go = (size_t)(t + 2) * 16 * ROWS;
      tdm_load_tile((unsigned)(pre2 * 1024 * 4), kp + go);
      tdm_load_tile((unsigned)((3072 + pre2 * 1024) * 4), vp + go);
    }
    const float* kt = smem + cur2 * 1024;
    const float* vt = smem + 3072 + cur2 * 1024;
    const int rs = 64;
    cur2 = (cur2 == 2) ? 0 : cur2 + 1;
    pre2 = (pre2 == 2) ? 0 : pre2 + 1;
#else
    const float* kt = kp + (size_t)t * 16 * ROWS;
    const float* vt = vp + (size_t)t * 16 * ROWS;
    const int rs = ROWS;
#endif

    v8f s = score_tile(kt, rs, n, hi, bQ);

    // p[j] = exp(clip(s - amax, -8, 0) - exp(-8)); lane-local qmax is this
    // lane's query stat (replicated in both halves).
    float p[8];
#pragma unroll
    for (int j = 0; j < 8; ++j) {
      float x = s[j] - qmax;
      x = fminf(fmaxf(x, -8.0f), 0.0f) - EXP_C;
      p[j] = __expf(x);
      halfsum += p[j];
    }

    // ---- V tile as B-fragments: bV[nn][c] covers keys 4c..4c+3,
    //      d-columns 16*nn + n ----
    v2f bV[4][4];
#pragma unroll
    for (int c = 0; c < 4; ++c) {
      const float* vr0 = vt + (4 * c + 2 * hi) * rs;
#pragma unroll
      for (int nn = 0; nn < 4; ++nn) {
        bV[nn][c].x = vr0[nn * 16 + n];
        bV[nn][c].y = vr0[rs + nn * 16 + n];
      }
    }

    // ---- rebuild A = P (query-major) from the D-layout S^T fragment.
    // P[qq][kk] lives in lane (qq + 16*(kk>>3)), register p[kk&7].
#pragma unroll
    for (int c = 0; c < 4; ++c) {
      const int kLo = 4 * c;         // key for lower-half lanes
      const int kHi = 4 * c + 2;     // key for upper-half lanes
      const int srcHalf = kLo >> 3;  // == kHi >> 3 for c in 0..3
      const int src = n + 16 * srcHalf;
      const float x_lo = __shfl(p[kLo & 7], src, 32);
      const float x_hi = __shfl(p[kHi & 7], src, 32);
      const float y_lo = __shfl(p[(kLo + 1) & 7], src, 32);
      const float y_hi = __shfl(p[(kHi + 1) & 7], src, 32);
      v2f aP;
      aP.x = hi ? x_hi : x_lo;
      aP.y = hi ? y_hi : y_lo;
      O0 = WMMA_F32(aP, bV[0][c], O0);
      O1 = WMMA_F32(aP, bV[1][c], O1);
      O2 = WMMA_F32(aP, bV[2][c], O2);
      O3 = WMMA_F32(aP, bV[3][c], O3);
    }
  }

  // ---- denominator clamp + exact reciprocal (per query, lane-resident) ----
  float sum = halfsum + __shfl_xor(halfsum, 16, 32);
  sum = fminf(fmaxf(sum, 1.0f - EXP_C), 1024.0f);
  const float recip = 1.0f / sum;

  // ---- scale rows of O (row m = j + 8*hi is a query) and store ----
#pragma unroll
  for (int j = 0; j < 8; ++j) {
    const float r_lo = __shfl(recip, j, 32);      // query j
    const float r_hi = __shfl(recip, j + 8, 32);  // query j+8
    const float r = hi ? r_hi : r_lo;
    const int m = j + 8 * hi;
    float* orow = op + (size_t)(qbase + m) * ROWS;
    orow[0 * 16 + n] = O0[j] * r;
    orow[1 * 16 + n] = O1[j] * r;
    orow[2 * 16 + n] = O2[j] * r;
    orow[3 * 16 + n] = O3[j] * r;
  }
}

extern "C" void kernel_launch(void* const* d_in, const int* in_sizes, int n_in,
                              void* d_out, int out_size, void* d_ws,
                              size_t ws_size, hipStream_t stream) {
  (void)in_sizes;
  (void)n_in;
  (void)out_size;
  (void)d_ws;
  (void)ws_size;
  const float* q = (const float*)d_in[0];
  const float* k = (const float*)d_in[1];
  const float* v = (const float*)d_in[2];
  float* out = (float*)d_out;

  // 4 waves per block, each wave owns a 16-query tile of the same (b,h).
  // 24 KB dynamic LDS: triple-buffered K and V tiles staged by the TDM.
  dim3 grid(Sn / 16 / 4, Hn, Bn);
  dim3 block(128, 1, 1);
  attn_fp32_wmma<<<grid, block, 24576, stream>>>(q, k, v, out);
}